// DotInteraction_8040178778582
// MI455X (gfx1250) — compile-verified
//
#include <hip/hip_runtime.h>

// DotInteraction: out[b, r*(r-1)/2 + c] = sum_d x[b,r,d]*x[b,c,d]  for r>c
// x: (16384, 32, 64) fp32;  out: (16384, 496) fp32
//
// One wave32 per batch. G = X*X^T computed as three 16x16 fp32 WMMA tiles
// (V_WMMA_F32_16X16X4_F32, K accumulated over 16 steps of 4).

typedef __attribute__((ext_vector_type(2))) float v2f;
typedef __attribute__((ext_vector_type(8))) float v8f;

#define NF   32
#define ND   64
#define NOUT 496   // 32*31/2

__global__ __launch_bounds__(256) void dot_interaction_wmma(
    const float* __restrict__ x, float* __restrict__ out, int nbatch)
{
    const int wave = threadIdx.x >> 5;
    const int lane = threadIdx.x & 31;
    const int b = blockIdx.x * 8 + wave;
    if (b >= nbatch) return;   // wave-uniform: EXEC stays all-1s for WMMA

    const float* xb = x + (size_t)b * (NF * ND);
    const int m    = lane & 15;   // M (and N) index within tile
    const int half = lane >> 4;   // selects K pair {0,1} vs {2,3}

    // A-fragment base pointers (fp32, 16x4 layout):
    //   lane m      : A[m][4k+0], A[m][4k+1]   in v0,v1
    //   lane m+16   : A[m][4k+2], A[m][4k+3]   in v0,v1
    const float* p0 = xb + m * ND + 2 * half;          // rows  0..15
    const float* p1 = xb + (16 + m) * ND + 2 * half;   // rows 16..31

    v8f acc00 = {0.f,0.f,0.f,0.f,0.f,0.f,0.f,0.f};  // G[0:16 , 0:16 ]
    v8f acc10 = acc00;                               // G[16:32, 0:16 ]
    v8f acc11 = acc00;                               // G[16:32, 16:32]

#pragma unroll
    for (int k = 0; k < 16; ++k) {
        v2f a0 = *(const v2f*)(p0 + 4 * k);   // global_load_b64
        v2f a1 = *(const v2f*)(p1 + 4 * k);   // global_load_b64
        // For G = X*X^T the B-operand (A_j^T) register layout equals the
        // A-fragment of A_j, so pass the same regs for both sides.
        acc00 = __builtin_amdgcn_wmma_f32_16x16x4_f32(
            false, a0, false, a0, (short)0, acc00, false, false);
        acc10 = __builtin_amdgcn_wmma_f32_16x16x4_f32(
            false, a1, false, a0, (short)0, acc10, false, false);
        acc11 = __builtin_amdgcn_wmma_f32_16x16x4_f32(
            false, a1, false, a1, (short)0, acc11, false, false);
    }

    // C/D layout: VGPR v, lanes 0-15 -> (M=v, N=m); lanes 16-31 -> (M=v+8, N=m)
    float* ob = out + (size_t)b * NOUT;
    const int c = m;
#pragma unroll
    for (int v = 0; v < 8; ++v) {
        const int rl = v + 8 * half;   // row within 16-row tile
        {   // tile (1,0): rows 16..31 x cols 0..15 — always strictly lower
            const int r = 16 + rl;
            ob[r * (r - 1) / 2 + c] = acc10[v];
        }
        if (rl > c) {   // tile (0,0): keep r > c
            const int r = rl;
            ob[r * (r - 1) / 2 + c] = acc00[v];
        }
        if (rl > c) {   // tile (1,1): rows/cols +16, keep r > c
            const int r  = 16 + rl;
            const int cc = 16 + c;
            ob[r * (r - 1) / 2 + cc] = acc11[v];
        }
    }
}

extern "C" void kernel_launch(void* const* d_in, const int* in_sizes, int n_in,
                              void* d_out, int out_size, void* d_ws, size_t ws_size,
                              hipStream_t stream) {
    const float* x = (const float*)d_in[0];
    float* out = (float*)d_out;
    const int nbatch = in_sizes[0] / (NF * ND);      // 16384
    const int nblocks = (nbatch + 7) / 8;            // 8 waves (batches) per block
    dot_interaction_wmma<<<nblocks, 256, 0, stream>>>(x, out, nbatch);
}